// AttentionPool_1760936591902
// MI455X (gfx1250) — compile-verified
//
#include <hip/hip_runtime.h>

// Problem constants (match reference)
#define B_    64
#define L_    4096
#define KD_   128
#define H_    4
#define OUT_  256

// Kernel-1 config: 4 waves/block, each wave owns one 16-row WMMA tile.
#define K1_WAVES        4
#define K1_THREADS      (K1_WAVES * 32)
#define ROWS_PER_WAVE   16
#define ROWS_PER_BLOCK  (K1_WAVES * ROWS_PER_WAVE)   // 64
#define CHUNKS          (L_ / ROWS_PER_BLOCK)        // 64
#define TILE_STRIDE     132                          // 128 + 4 pad floats -> conflict-free ds b64/b128
#define PART_FLOATS     (8 + KD_ * H_)               // m[4], Z[4], acc[512]

typedef __attribute__((ext_vector_type(2))) float v2f;
typedef __attribute__((ext_vector_type(4))) float v4f;
typedef __attribute__((ext_vector_type(8))) float v8f;

// ---------------------------------------------------------------------------
// Kernel 1: stream x once (async DMA into LDS); WMMA-scored online softmax;
// per-block partials:  partial[b][c] = { m[4], Z[4], acc[k*4+h] (512) }
// ---------------------------------------------------------------------------
__global__ __launch_bounds__(K1_THREADS)
void pool_partial_kernel(const float* __restrict__ x,
                         const unsigned char* __restrict__ mask,
                         const float* __restrict__ queries,
                         float* __restrict__ partial)
{
    __shared__ float qT[KD_ * 16];                                   // qT[k*16+n], n>=4 zero
    __shared__ float tile[K1_WAVES][ROWS_PER_WAVE][TILE_STRIDE];     // 33,792 B
    __shared__ float pbuf[K1_WAVES][ROWS_PER_WAVE][H_];
    __shared__ unsigned char mbuf[ROWS_PER_BLOCK];
    __shared__ float hdr[K1_WAVES][8];                               // m[4], Z[4] per wave
    __shared__ float accw[K1_WAVES][KD_ * H_];
    __shared__ float Msh[H_];
    __shared__ float ssc[K1_WAVES][H_];

    const int b    = blockIdx.y;
    const int c    = blockIdx.x;
    const int tid  = threadIdx.x;
    const int w    = tid >> 5;
    const int lane = tid & 31;
    const int rowBase = c * ROWS_PER_BLOCK;

    // ---- async DMA: memory -> LDS, no VGPR staging (ASYNCcnt-tracked) ----
    // Block-uniform base in SGPRs; wave/row/lane component as 32-bit VGPR offset.
    const float* xblock = x + ((size_t)b * L_ + rowBase) * KD_;
    #pragma unroll
    for (int r = 0; r < ROWS_PER_WAVE; ++r) {
        const unsigned voffs = ((unsigned)((w * ROWS_PER_WAVE + r) * KD_) + lane * 4) * 4u;
        const unsigned ldsb  = (unsigned)(uintptr_t)&tile[w][r][lane * 4];
        asm volatile("global_load_async_to_lds_b128 %0, %1, %2 th:TH_LOAD_NT"
                     :: "v"(ldsb), "v"(voffs), "s"(xblock)
                     : "memory");
    }

    // Overlap with the DMA: build zero-padded q^T and load mask bytes.
    for (int i = tid; i < KD_ * 16; i += K1_THREADS) {
        const int k = i >> 4, n = i & 15;
        qT[i] = (n < H_) ? queries[n * KD_ + k] : 0.0f;
    }
    if (tid < ROWS_PER_BLOCK)
        mbuf[tid] = mask[(size_t)b * L_ + rowBase + tid];

    asm volatile("s_wait_asynccnt 0x0" ::: "memory");
    __syncthreads();

    // ---- scores via V_WMMA_F32_16X16X4_F32: D[16 rows][16 cols] (cols 0..3 = heads)
    // Two independent accumulators to break the D->C dependency chain.
    const int am = lane & 15;     // A row / B,N column index
    const int kh = lane >> 4;     // which K pair this half-wave holds
    v8f cacc0 = {}, cacc1 = {};
    for (int k0 = 0; k0 < KD_; k0 += 8) {
        {
            const v2f a = *(const v2f*)&tile[w][am][k0 + 2 * kh];
            v2f bq;
            bq.x = qT[(k0     + 2 * kh) * 16 + am];
            bq.y = qT[(k0 + 1 + 2 * kh) * 16 + am];
            cacc0 = __builtin_amdgcn_wmma_f32_16x16x4_f32(
                false, a, false, bq, (short)0, cacc0, false, false);
        }
        {
            const int k1 = k0 + 4;
            const v2f a = *(const v2f*)&tile[w][am][k1 + 2 * kh];
            v2f bq;
            bq.x = qT[(k1     + 2 * kh) * 16 + am];
            bq.y = qT[(k1 + 1 + 2 * kh) * 16 + am];
            cacc1 = __builtin_amdgcn_wmma_f32_16x16x4_f32(
                false, a, false, bq, (short)0, cacc1, false, false);
        }
    }

    // Lane holds column n=am; rows rr = v + 8*kh. Apply mask, local softmax stats.
    float s[8];
    #pragma unroll
    for (int v = 0; v < 8; ++v) {
        const int rr = v + 8 * kh;
        float sv = cacc0[v] + cacc1[v];
        if (!mbuf[w * ROWS_PER_WAVE + rr]) sv = -1e9f;
        s[v] = sv;
    }
    float mloc = s[0];
    #pragma unroll
    for (int v = 1; v < 8; ++v) mloc = fmaxf(mloc, s[v]);
    const float mh = fmaxf(mloc, __shfl_xor(mloc, 16, 32));    // max over 16 rows (per column)

    float p[8];
    float zloc = 0.0f;
    #pragma unroll
    for (int v = 0; v < 8; ++v) { p[v] = __expf(s[v] - mh); zloc += p[v]; }
    const float Zh = zloc + __shfl_xor(zloc, 16, 32);

    if (am < H_) {
        #pragma unroll
        for (int v = 0; v < 8; ++v) pbuf[w][v + 8 * kh][am] = p[v];
        if (kh == 0) { hdr[w][am] = mh; hdr[w][4 + am] = Zh; }
    }
    __syncthreads();

    // ---- pooled accumulation: lane owns k-slice [4*lane, 4*lane+3], all 4 heads ----
    v4f acch[H_];
    #pragma unroll
    for (int h = 0; h < H_; ++h) acch[h] = (v4f){0.f, 0.f, 0.f, 0.f};
    #pragma unroll
    for (int r = 0; r < ROWS_PER_WAVE; ++r) {
        const v4f xr = *(const v4f*)&tile[w][r][lane * 4];      // conflict-free b128
        const v4f pv = *(const v4f*)&pbuf[w][r][0];             // broadcast
        acch[0] += xr * pv.x;
        acch[1] += xr * pv.y;
        acch[2] += xr * pv.z;
        acch[3] += xr * pv.w;
    }
    #pragma unroll
    for (int j = 0; j < 4; ++j) {
        const int kk = lane * 4 + j;
        #pragma unroll
        for (int h = 0; h < H_; ++h) accw[w][kk * H_ + h] = acch[h][j];
    }
    __syncthreads();

    // ---- merge the 4 wave-partials (log-sum-exp) and emit block partial ----
    if (tid < H_) {
        float m = hdr[0][tid];
        #pragma unroll
        for (int ww = 1; ww < K1_WAVES; ++ww) m = fmaxf(m, hdr[ww][tid]);
        Msh[tid] = m;
    }
    __syncthreads();
    if (tid < K1_WAVES * H_)
        ssc[tid >> 2][tid & 3] = __expf(hdr[tid >> 2][tid & 3] - Msh[tid & 3]);
    __syncthreads();

    float* outp = partial + ((size_t)b * CHUNKS + c) * PART_FLOATS;
    if (tid < H_) {
        outp[tid] = Msh[tid];
    } else if (tid < 2 * H_) {
        const int h = tid - H_;
        float z = 0.0f;
        #pragma unroll
        for (int ww = 0; ww < K1_WAVES; ++ww) z += hdr[ww][4 + h] * ssc[ww][h];
        outp[tid] = z;
    }
    for (int i = tid; i < KD_ * H_; i += K1_THREADS) {
        const int h = i & 3;
        float v = 0.0f;
        #pragma unroll
        for (int ww = 0; ww < K1_WAVES; ++ww) v += accw[ww][i] * ssc[ww][h];
        outp[8 + i] = v;
    }
}

// ---------------------------------------------------------------------------
// Kernel 2: merge the CHUNKS partials per batch -> normalized pooled[b][512]
// ---------------------------------------------------------------------------
__global__ __launch_bounds__(128)
void merge_kernel(const float* __restrict__ partial, float* __restrict__ pooled)
{
    __shared__ float hdr[CHUNKS][8];
    __shared__ float sc[CHUNKS][H_];
    __shared__ float Msh[H_];
    __shared__ float Zsh[H_];

    const int b = blockIdx.x;
    const int tid = threadIdx.x;
    const float* pb = partial + (size_t)b * CHUNKS * PART_FLOATS;

    for (int i = tid; i < CHUNKS * 8; i += 128)
        hdr[i >> 3][i & 7] = pb[(size_t)(i >> 3) * PART_FLOATS + (i & 7)];
    __syncthreads();

    if (tid < H_) {
        float m = hdr[0][tid];
        for (int c = 1; c < CHUNKS; ++c) m = fmaxf(m, hdr[c][tid]);
        Msh[tid] = m;
    }
    __syncthreads();
    for (int i = tid; i < CHUNKS * H_; i += 128)
        sc[i >> 2][i & 3] = __expf(hdr[i >> 2][i & 3] - Msh[i & 3]);
    __syncthreads();
    if (tid < H_) {
        float z = 0.0f;
        for (int c = 0; c < CHUNKS; ++c) z += hdr[c][4 + tid] * sc[c][tid];
        Zsh[tid] = z;
    }
    __syncthreads();

    for (int i = tid; i < KD_ * H_; i += 128) {
        const int h = i & 3;
        float v = 0.0f;
        for (int c = 0; c < CHUNKS; ++c)
            v += pb[(size_t)c * PART_FLOATS + 8 + i] * sc[c][h];
        pooled[(size_t)b * (KD_ * H_) + i] = v / Zsh[h];
    }
}

// ---------------------------------------------------------------------------
// Kernel 3: out = relu(pooled @ W^T + bias + relu(emb[num]))  via WMMA f32
//   M=64, N=256, K=512; one wave per 16x16 tile -> 64 waves = 8 blocks.
//   4 independent accumulators: only 64 waves exist, so the WMMA dependency
//   chain is the latency limiter -- interleave 4 chains of 32.
// ---------------------------------------------------------------------------
__global__ __launch_bounds__(256)
void gemm_out_kernel(const float* __restrict__ pooled,
                     const float* __restrict__ W,
                     const float* __restrict__ bias,
                     const float* __restrict__ emb,
                     const int* __restrict__ num,
                     float* __restrict__ out)
{
    const int tid  = threadIdx.x;
    const int w    = tid >> 5;
    const int lane = tid & 31;
    const int wg   = blockIdx.x * 8 + w;          // 0..63
    const int m0   = (wg >> 4) * 16;              // batch tile
    const int n0   = (wg & 15) * 16;              // OUT tile
    const int am   = lane & 15;
    const int kh   = lane >> 4;

    const float* arow = pooled + (size_t)(m0 + am) * (KD_ * H_);
    const float* brow = W + (size_t)(n0 + am) * (KD_ * H_);   // B[k][n] = W[n][k]

    v8f c0 = {}, c1 = {}, c2 = {}, c3 = {};
    for (int k0 = 0; k0 < KD_ * H_; k0 += 16) {
        {
            const v2f a  = *(const v2f*)(arow + k0 + 2 * kh);
            const v2f bb = *(const v2f*)(brow + k0 + 2 * kh);
            c0 = __builtin_amdgcn_wmma_f32_16x16x4_f32(false, a, false, bb, (short)0, c0, false, false);
        }
        {
            const v2f a  = *(const v2f*)(arow + k0 + 4 + 2 * kh);
            const v2f bb = *(const v2f*)(brow + k0 + 4 + 2 * kh);
            c1 = __builtin_amdgcn_wmma_f32_16x16x4_f32(false, a, false, bb, (short)0, c1, false, false);
        }
        {
            const v2f a  = *(const v2f*)(arow + k0 + 8 + 2 * kh);
            const v2f bb = *(const v2f*)(brow + k0 + 8 + 2 * kh);
            c2 = __builtin_amdgcn_wmma_f32_16x16x4_f32(false, a, false, bb, (short)0, c2, false, false);
        }
        {
            const v2f a  = *(const v2f*)(arow + k0 + 12 + 2 * kh);
            const v2f bb = *(const v2f*)(brow + k0 + 12 + 2 * kh);
            c3 = __builtin_amdgcn_wmma_f32_16x16x4_f32(false, a, false, bb, (short)0, c3, false, false);
        }
    }
    const v8f c = (c0 + c1) + (c2 + c3);

    #pragma unroll
    for (int v = 0; v < 8; ++v) {
        const int m = m0 + v + 8 * kh;
        const int n = n0 + am;
        const float e = emb[(size_t)num[m] * OUT_ + n];
        const float val = c[v] + bias[n] + fmaxf(e, 0.0f);
        out[(size_t)m * OUT_ + n] = fmaxf(val, 0.0f);
    }
}

// ---------------------------------------------------------------------------
extern "C" void kernel_launch(void* const* d_in, const int* in_sizes, int n_in,
                              void* d_out, int out_size, void* d_ws, size_t ws_size,
                              hipStream_t stream) {
    (void)in_sizes; (void)n_in; (void)out_size; (void)ws_size;

    const float*         x       = (const float*)d_in[0];
    const unsigned char* mask    = (const unsigned char*)d_in[1];  // jnp.bool_ -> 1 byte
    const int*           num     = (const int*)d_in[2];
    const float*         queries = (const float*)d_in[3];
    const float*         W       = (const float*)d_in[4];
    const float*         bias    = (const float*)d_in[5];
    const float*         emb     = (const float*)d_in[6];
    float*               out     = (float*)d_out;

    float* partial = (float*)d_ws;                                   // 64*64*520 f32 ~ 8.5 MB
    float* pooled  = partial + (size_t)B_ * CHUNKS * PART_FLOATS;    // + 64*512 f32

    pool_partial_kernel<<<dim3(CHUNKS, B_), K1_THREADS, 0, stream>>>(x, mask, queries, partial);
    merge_kernel<<<dim3(B_), 128, 0, stream>>>(partial, pooled);
    gemm_out_kernel<<<dim3(8), 256, 0, stream>>>(pooled, W, bias, emb, num, out);
}